// DiGCN_Inception_Block_Ranking_83202106458340
// MI455X (gfx1250) — compile-verified
//
#include <hip/hip_runtime.h>

typedef __attribute__((ext_vector_type(2))) float v2f;
typedef __attribute__((ext_vector_type(8))) float v8f;

// ---------------------------------------------------------------------------
// Fused triple GEMM:  O0 = X*W0 + (B0+B1+B2),  O1 = X*W1,  O2 = X*W2
// X: [nrows x K] row-major, W*: [K x DOUT] row-major.
// One wave per 16-row strip; native fp32 WMMA 16x16x4, K stepped by 4.
// A frag (16x4):  lanes 0-15 -> M=lane, v0/v1 = K0/K1; lanes 16-31 -> K2/K3.
// B frag (4x16):  lanes 0-15 -> N=lane, v0/v1 = K0/K1; lanes 16-31 -> K2/K3.
// D (16x16): vgpr r -> row r + 8*(lane>=16), col = lane&15.
// ---------------------------------------------------------------------------
template <int K, int DOUT>
__global__ __launch_bounds__(256) void gemm3_wmma(
    const float* __restrict__ X,
    const float* __restrict__ W0, const float* __restrict__ W1,
    const float* __restrict__ W2,
    const float* __restrict__ B0, const float* __restrict__ B1,
    const float* __restrict__ B2,
    float* __restrict__ O0, float* __restrict__ O1, float* __restrict__ O2,
    int nrows)
{
    constexpr int NT = DOUT / 16;               // 16-wide output tiles
    const int wave = (int)((blockIdx.x * blockDim.x + threadIdx.x) >> 5);
    const int lane = threadIdx.x & 31;
    const int row0 = wave * 16;
    if (row0 >= nrows) return;                  // wave-uniform exit (EXEC stays full)

    const int lo = lane & 15;                   // M (for A) / N (for B,D)
    const int hi = lane >> 4;                   // selects K pair {0,1} vs {2,3}

    v8f acc0[NT], acc1[NT], acc2[NT];
#pragma unroll
    for (int t = 0; t < NT; ++t) {
        acc0[t] = (v8f){0.f,0.f,0.f,0.f,0.f,0.f,0.f,0.f};
        acc1[t] = (v8f){0.f,0.f,0.f,0.f,0.f,0.f,0.f,0.f};
        acc2[t] = (v8f){0.f,0.f,0.f,0.f,0.f,0.f,0.f,0.f};
    }

    const float* xrow = X + (size_t)(row0 + lo) * K;

    for (int k0 = 0; k0 < K; k0 += 4) {
        v2f a;
        a.x = xrow[k0 + 2 * hi];
        a.y = xrow[k0 + 2 * hi + 1];
#pragma unroll
        for (int t = 0; t < NT; ++t) {
            const int boff = (k0 + 2 * hi) * DOUT + t * 16 + lo;
            v2f b0, b1, b2;
            b0.x = W0[boff]; b0.y = W0[boff + DOUT];
            b1.x = W1[boff]; b1.y = W1[boff + DOUT];
            b2.x = W2[boff]; b2.y = W2[boff + DOUT];
            acc0[t] = __builtin_amdgcn_wmma_f32_16x16x4_f32(
                false, a, false, b0, (short)0, acc0[t], false, false);
            acc1[t] = __builtin_amdgcn_wmma_f32_16x16x4_f32(
                false, a, false, b1, (short)0, acc1[t], false, false);
            acc2[t] = __builtin_amdgcn_wmma_f32_16x16x4_f32(
                false, a, false, b2, (short)0, acc2[t], false, false);
        }
    }

#pragma unroll
    for (int t = 0; t < NT; ++t) {
        const int col = t * 16 + lo;
        const float bias = B0[col] + B1[col] + B2[col];
#pragma unroll
        for (int r = 0; r < 8; ++r) {
            const size_t off = (size_t)(row0 + r + 8 * hi) * DOUT + col;
            O0[off] = acc0[t][r] + bias;
            O1[off] = acc1[t][r];
            O2[off] = acc2[t][r];
        }
    }
}

// ---------------------------------------------------------------------------
// Fused edge aggregation for both conv branches of one inception block:
//   out[dst[e]*D + d] += ew1[e]*h1[src[e]*D + d] + ew2[e]*h2[src[e]*D + d]
// One thread per (edge, feature-dim). h1/h2/out live in L2 (25.6 MB each).
// ---------------------------------------------------------------------------
template <int LOGD>
__global__ __launch_bounds__(256) void edge_agg(
    const int* __restrict__ src, const int* __restrict__ dst,
    const float* __restrict__ ew1, const float* __restrict__ ew2,
    const float* __restrict__ h1, const float* __restrict__ h2,
    float* __restrict__ out, long long total)
{
    long long i = (long long)blockIdx.x * blockDim.x + threadIdx.x;
    const long long stride = (long long)gridDim.x * blockDim.x;
    for (; i < total; i += stride) {
        const int e = (int)(i >> LOGD);
        const int d = (int)(i & ((1 << LOGD) - 1));
        const int s = src[e];
        const int t = dst[e];
        const size_t so = ((size_t)s << LOGD) + d;
        const float v = ew1[e] * h1[so] + ew2[e] * h2[so];
        (void)__hip_atomic_fetch_add(out + (((size_t)t << LOGD) + d), v,
                                     __ATOMIC_RELAXED, __HIP_MEMORY_SCOPE_AGENT);
    }
}

extern "C" void kernel_launch(void* const* d_in, const int* in_sizes, int n_in,
                              void* d_out, int out_size, void* d_ws, size_t ws_size,
                              hipStream_t stream)
{
    const float* features = (const float*)d_in[0];
    const float* ew1      = (const float*)d_in[1];
    const float* ew2      = (const float*)d_in[2];
    const int*   src      = (const int*)  d_in[3];
    const int*   dst      = (const int*)  d_in[4];
    const float* ln1_w = (const float*)d_in[5];  const float* ln1_b = (const float*)d_in[6];
    const float* c11_w = (const float*)d_in[7];  const float* c11_b = (const float*)d_in[8];
    const float* c12_w = (const float*)d_in[9];  const float* c12_b = (const float*)d_in[10];
    const float* ln2_w = (const float*)d_in[11]; const float* ln2_b = (const float*)d_in[12];
    const float* c21_w = (const float*)d_in[13]; const float* c21_b = (const float*)d_in[14];
    const float* c22_w = (const float*)d_in[15]; const float* c22_b = (const float*)d_in[16];

    constexpr int FIN = 128, EMB = 64, OUT = 32;
    const int N = in_sizes[0] / FIN;
    const int E = in_sizes[1];

    // Workspace layout: [xsum N*64][h1 N*64][h2 N*64]; g1/g2 overlay h1 region.
    float* xsum = (float*)d_ws;
    float* h1   = xsum + (size_t)N * EMB;
    float* h2   = h1   + (size_t)N * EMB;
    float* g1   = h1;                      // h1/h2 dead after block-1 edge pass
    float* g2   = g1   + (size_t)N * OUT;
    float* outp = (float*)d_out;

    const int threads = 256;
    const int waves   = (N + 15) / 16;
    const int gblocks = (waves * 32 + threads - 1) / threads;

    // Block 1: x0(+all biases) -> xsum, h1 = f@c11_w, h2 = f@c12_w
    gemm3_wmma<FIN, EMB><<<gblocks, threads, 0, stream>>>(
        features, ln1_w, c11_w, c12_w, ln1_b, c11_b, c12_b,
        xsum, h1, h2, N);

    // Block 1 edge aggregation into xsum
    {
        const long long total = (long long)E * EMB;
        const int blocks = (int)((total + threads - 1) / threads);
        edge_agg<6><<<blocks, threads, 0, stream>>>(
            src, dst, ew1, ew2, h1, h2, xsum, total);
    }

    // Block 2: y0(+all biases) -> d_out, g1 = x@c21_w, g2 = x@c22_w
    gemm3_wmma<EMB, OUT><<<gblocks, threads, 0, stream>>>(
        xsum, ln2_w, c21_w, c22_w, ln2_b, c21_b, c22_b,
        outp, g1, g2, N);

    // Block 2 edge aggregation into d_out
    {
        const long long total = (long long)E * OUT;
        const int blocks = (int)((total + threads - 1) / threads);
        edge_agg<5><<<blocks, threads, 0, stream>>>(
            src, dst, ew1, ew2, g1, g2, outp, total);
    }
}